// Qwen2Attention_33423435497498
// MI455X (gfx1250) — compile-verified
//
#include <hip/hip_runtime.h>
#include <hip/hip_bf16.h>
#include <math.h>

// ---------------- Problem constants (from reference) ----------------
#define HIDDEN   2048
#define NH       16
#define NKV      4
#define HD       128       // head_dim
#define D2       64        // head_dim / 2 (rope half)
#define QSIZE    2048
#define KVSIZE   512
#define QKVLD    3072      // qkv row stride (Q_SIZE + 2*KV_SIZE)
#define KOFF     2048      // K column offset in qkv
#define VOFF     2560      // V column offset in qkv
#define SEQB     2048      // SEQ
#define BATCH    2
#define NTOK     (BATCH * SEQB)
#define SM_SCALE 0.08838834764831845f   // 1/sqrt(128)

typedef __attribute__((ext_vector_type(2))) float v2f;
typedef __attribute__((ext_vector_type(8))) float v8f;

// Exact-f32 tensor op: D = A(16x4) * B(4x16) + C(16x16)
__device__ __forceinline__ v8f wmma_f32(v2f a, v2f b, v8f c) {
  return __builtin_amdgcn_wmma_f32_16x16x4_f32(
      /*neg_a=*/false, a, /*neg_b=*/false, b,
      /*c_mod=*/(short)0, c, /*reuse_a=*/false, /*reuse_b=*/false);
}

// =====================================================================
// GEMM (NT):  C[M,N] = A[M,K] * B[N,K]^T (+ bias[N])
// Block: 128 threads = 4 waves. Tile 128(M) x 64(N), BK = 32.
// Wave w owns rows [32w, 32w+32) as two 16-row stripes; every B fragment
// feeds two WMMAs (0.75 DS b64 loads per WMMA).
// =====================================================================
#define GBK  32
#define GLDP 36   // LDS row stride (32 + 4 pad): 8B aligned, conflict-free frags

__global__ __launch_bounds__(128)
void gemm_nt_kernel(const float* __restrict__ A, const float* __restrict__ Bw,
                    const float* __restrict__ bias, float* __restrict__ C,
                    int M, int N, int K) {
  __shared__ float sA[128 * GLDP];
  __shared__ float sB[64 * GLDP];
  const int t    = threadIdx.x;
  const int lane = t & 31, wid = t >> 5;
  const int hi   = lane >> 4, l15 = lane & 15;
  const int mbase = blockIdx.y * 128;
  const int nbase = blockIdx.x * 64;

  const v8f vz = {0.f,0.f,0.f,0.f,0.f,0.f,0.f,0.f};
  v8f acc[2][4] = {{vz, vz, vz, vz}, {vz, vz, vz, vz}};

  // staging: rows hold 8 float4s; thread t handles f4 index t+128*i
  const int srow = t >> 3;          // 0..15 base row group
  const int scol = (t & 7) << 2;    // 0..28

  for (int kb = 0; kb < K; kb += GBK) {
    // A tile: 128 rows x 32 cols = 1024 f4 -> 8 per thread
#pragma unroll
    for (int i = 0; i < 8; i++) {
      const int row = srow + 16 * i;
      *(float4*)&sA[row * GLDP + scol] =
          *(const float4*)&A[(size_t)(mbase + row) * K + kb + scol];
    }
    // B tile: 64 rows x 32 cols = 512 f4 -> 4 per thread
#pragma unroll
    for (int i = 0; i < 4; i++) {
      const int row = srow + 16 * i;
      *(float4*)&sB[row * GLDP + scol] =
          *(const float4*)&Bw[(size_t)(nbase + row) * K + kb + scol];
    }
    if (kb + GBK < K) {  // hint next chunk into cache (global_prefetch_b8)
      __builtin_prefetch(&A[(size_t)(mbase + srow) * K + kb + GBK + scol], 0, 0);
      __builtin_prefetch(&Bw[(size_t)(nbase + srow) * K + kb + GBK + scol], 0, 0);
    }
    __syncthreads();
#pragma unroll
    for (int k2 = 0; k2 < GBK; k2 += 4) {
      const v2f a0 = *(const v2f*)&sA[(wid * 32 + l15) * GLDP + k2 + 2 * hi];
      const v2f a1 = *(const v2f*)&sA[(wid * 32 + 16 + l15) * GLDP + k2 + 2 * hi];
#pragma unroll
      for (int nc = 0; nc < 4; nc++) {
        const v2f b = *(const v2f*)&sB[(nc * 16 + l15) * GLDP + k2 + 2 * hi];
        acc[0][nc] = wmma_f32(a0, b, acc[0][nc]);
        acc[1][nc] = wmma_f32(a1, b, acc[1][nc]);
      }
    }
    __syncthreads();
  }

#pragma unroll
  for (int st = 0; st < 2; st++) {
#pragma unroll
    for (int nc = 0; nc < 4; nc++) {
      const int n  = nbase + nc * 16 + l15;
      const float bv = bias ? bias[n] : 0.0f;
#pragma unroll
      for (int r = 0; r < 8; r++) {
        const int m = mbase + wid * 32 + st * 16 + r + 8 * hi;
        C[(size_t)m * N + n] = acc[st][nc][r] + bv;
      }
    }
  }
}

// =====================================================================
// RoPE (in-place on Q and K parts of qkv). One thread per (token, head, d<64).
// heads 0..15 -> Q, heads 16..19 -> K. 4096*20*64 = 5,242,880 threads.
// =====================================================================
__global__ __launch_bounds__(256)
void rope_kernel(float* __restrict__ qkv) {
  const int tid   = blockIdx.x * 256 + threadIdx.x;
  const int token = tid / 1280;
  const int rem   = tid % 1280;
  const int head  = rem >> 6;      // 0..19
  const int d     = rem & 63;      // 0..63
  const int pos   = token & (SEQB - 1);
  const int col   = (head < NH) ? head * HD + d : QSIZE + (head - NH) * HD + d;
  const float inv_freq = powf(1.0e6f, -(float)d * (1.0f / 64.0f));
  float s, c;
  sincosf((float)pos * inv_freq, &s, &c);
  float* p = qkv + (size_t)token * QKVLD + col;
  const float x1 = p[0], x2 = p[D2];
  p[0]  = x1 * c - x2 * s;
  p[D2] = x1 * s + x2 * c;
}

// =====================================================================
// Flash attention. Grid (32 qblocks, 16 heads, 2 batch), 128 threads.
// Each wave owns a 16-row q stripe; streams 16-key KV blocks with online
// softmax; P re-layout via per-wave LDS tile; O accumulated in 8 v8f regs.
// =====================================================================
__global__ __launch_bounds__(128)
void flash_attn_kernel(const float* __restrict__ qkv, float* __restrict__ out) {
  __shared__ float sQ[64 * 132];        // 64 q-rows x 128 dims (+4 pad)
  __shared__ float sK[16 * 132];        // 16 keys x 128 dims (+4 pad)
  __shared__ float sVt[128 * 20];       // transposed V: [dim][key] (+4 pad)
  __shared__ float sP[4][16 * 18];      // per-wave 16x16 P tile (+2 pad)

  const int t    = threadIdx.x;
  const int lane = t & 31, wid = t >> 5;
  const int hi   = lane >> 4, l15 = lane & 15;
  const int qblk = blockIdx.x;
  const int h    = blockIdx.y;
  const int b    = blockIdx.z;
  const int kvh  = h >> 2;                  // GQA: head h uses kv head h/4
  const int qbase = qblk * 64;
  const size_t rowbase = (size_t)(b * SEQB) * QKVLD;

  // ---- stage Q (pre-scaled by 1/sqrt(d)) ----
#pragma unroll
  for (int i = 0; i < 16; i++) {
    const int idx = t + 128 * i;           // 0..2047 float4s
    const int row = idx >> 5, c4 = (idx & 31) << 2;
    float4 v = *(const float4*)&qkv[rowbase + (size_t)(qbase + row) * QKVLD + h * HD + c4];
    v.x *= SM_SCALE; v.y *= SM_SCALE; v.z *= SM_SCALE; v.w *= SM_SCALE;
    *(float4*)&sQ[row * 132 + c4] = v;
  }

  const v8f vz = {0.f,0.f,0.f,0.f,0.f,0.f,0.f,0.f};
  v8f accO[8] = {vz, vz, vz, vz, vz, vz, vz, vz};
  float rm[8], rs[8];
#pragma unroll
  for (int r = 0; r < 8; r++) { rm[r] = -__builtin_inff(); rs[r] = 0.0f; }

  __syncthreads();

  const int nkb = (qbase >> 4) + 4;        // kv blocks up to & incl. diagonal
  for (int kbi = 0; kbi < nkb; kbi++) {
    const int kb = kbi * 16;
    // ---- stage K block (16 x 128, row-major) ----
#pragma unroll
    for (int i = 0; i < 4; i++) {
      const int idx = t + 128 * i;         // 0..511 float4s
      const int row = idx >> 5, c4 = (idx & 31) << 2;
      const float4 v = *(const float4*)&qkv[rowbase + (size_t)(kb + row) * QKVLD + KOFF + kvh * HD + c4];
      *(float4*)&sK[row * 132 + c4] = v;
    }
    // ---- stage V block transposed: sVt[dim][key] ----
#pragma unroll
    for (int i = 0; i < 16; i++) {
      sVt[t * 20 + i] = qkv[rowbase + (size_t)(kb + i) * QKVLD + VOFF + kvh * HD + t];
    }
    __syncthreads();

    // ---- S = Qstripe (16x128) * K^T (128x16) ----
    v8f s = vz;
#pragma unroll
    for (int k0 = 0; k0 < HD; k0 += 4) {
      const v2f a  = *(const v2f*)&sQ[(wid * 16 + l15) * 132 + k0 + 2 * hi];
      const v2f bb = *(const v2f*)&sK[l15 * 132 + k0 + 2 * hi];
      s = wmma_f32(a, bb, s);
    }

    // ---- causal mask + online softmax (row = r + 8*hi across 16 lanes) ----
    const int kcol = kb + l15;
#pragma unroll
    for (int r = 0; r < 8; r++) {
      const int mrow = qbase + wid * 16 + r + 8 * hi;
      const float sv = (kcol <= mrow) ? s[r] : -__builtin_inff();
      float mx = sv;
      mx = fmaxf(mx, __shfl_xor(mx, 1, 32));
      mx = fmaxf(mx, __shfl_xor(mx, 2, 32));
      mx = fmaxf(mx, __shfl_xor(mx, 4, 32));
      mx = fmaxf(mx, __shfl_xor(mx, 8, 32));
      const float newm  = fmaxf(rm[r], mx);
      const float alpha = __expf(rm[r] - newm);   // 0 on first block
      const float pv    = __expf(sv - newm);
      float ps = pv;
      ps += __shfl_xor(ps, 1, 32);
      ps += __shfl_xor(ps, 2, 32);
      ps += __shfl_xor(ps, 4, 32);
      ps += __shfl_xor(ps, 8, 32);
      rs[r] = rs[r] * alpha + ps;
      rm[r] = newm;
#pragma unroll
      for (int nc = 0; nc < 8; nc++) accO[nc][r] *= alpha;
      sP[wid][(r + 8 * hi) * 18 + l15] = pv;      // C-layout -> LDS
    }
    __syncthreads();   // make P visible across lanes (and fence LDS)

    // ---- O += P (16x16) * V (16x128) ----
#pragma unroll
    for (int k0 = 0; k0 < 16; k0 += 4) {
      const v2f a = *(const v2f*)&sP[wid][l15 * 18 + k0 + 2 * hi];
#pragma unroll
      for (int nc = 0; nc < 8; nc++) {
        const v2f bb = *(const v2f*)&sVt[(nc * 16 + l15) * 20 + k0 + 2 * hi];
        accO[nc] = wmma_f32(a, bb, accO[nc]);
      }
    }
    __syncthreads();   // before restaging sK/sVt
  }

  // ---- normalize + write [token][h*128 + d] ----
#pragma unroll
  for (int nc = 0; nc < 8; nc++) {
#pragma unroll
    for (int r = 0; r < 8; r++) {
      const int mrow = qbase + wid * 16 + r + 8 * hi;
      out[(size_t)(b * SEQB + mrow) * HIDDEN + h * HD + nc * 16 + l15] = accO[nc][r] / rs[r];
    }
  }
}

// =====================================================================
extern "C" void kernel_launch(void* const* d_in, const int* in_sizes, int n_in,
                              void* d_out, int out_size, void* d_ws, size_t ws_size,
                              hipStream_t stream) {
  const float* hs    = (const float*)d_in[0];   // (4096, 2048)
  const float* qkv_w = (const float*)d_in[1];   // (3072, 2048)
  const float* qkv_b = (const float*)d_in[2];   // (3072,)
  const float* o_w   = (const float*)d_in[3];   // (2048, 2048)
  // d_in[4] position_ids (reference uses arange internally), d_in[5] cu_seqlens: unused
  float* out  = (float*)d_out;
  float* qkv  = (float*)d_ws;                          // 4096 x 3072
  float* attn = qkv + (size_t)NTOK * QKVLD;            // 4096 x 2048

  // 1) QKV = hs * qkv_w^T + b
  dim3 g1(QKVLD / 64, NTOK / 128);
  gemm_nt_kernel<<<g1, 128, 0, stream>>>(hs, qkv_w, qkv_b, qkv, NTOK, QKVLD, HIDDEN);

  // 2) RoPE in-place on Q and K
  rope_kernel<<<(NTOK * 1280) / 256, 256, 0, stream>>>(qkv);

  // 3) causal GQA attention
  dim3 g3(SEQB / 64, NH, BATCH);
  flash_attn_kernel<<<g3, 128, 0, stream>>>(qkv, attn);

  // 4) out = attn * o_w^T
  dim3 g4(HIDDEN / 64, NTOK / 128);
  gemm_nt_kernel<<<g4, 128, 0, stream>>>(attn, o_w, nullptr, out, NTOK, HIDDEN, HIDDEN);
}